// L_spa_88115549045530
// MI455X (gfx1250) — compile-verified
//
#include <hip/hip_runtime.h>
#include <cstdint>
#include <cstddef>

// ---------------------------------------------------------------------------
// MI455X (gfx1250) plan:
//   pool_diff_kernel : streaming channel-mean + 4x4x4 avg-pool difference,
//                      fed by CDNA5 async global->LDS copies (ASYNCcnt),
//                      double-buffered 8-row stages per wave.
//   stencil_kernel   : 6-neighbor zero-padded (g - shift)^2 sum on the tiny
//                      (4,32,32,32) pooled volume.
// Bandwidth-bound: 201 MB read -> ~8.6 us floor at 23.3 TB/s.
// ---------------------------------------------------------------------------

#ifndef __has_builtin
#define __has_builtin(x) 0
#endif

#if __has_builtin(__builtin_amdgcn_global_load_async_to_lds_b128)
#define HAVE_ASYNC_LDS 1
#else
#define HAVE_ASYNC_LDS 0
#endif

#if HAVE_ASYNC_LDS
// Builtin signature (from clang diagnostic): 
//   void __builtin_amdgcn_global_load_async_to_lds_b128(
//       as1 v4i* gsrc, as3 v4i* ldst, imm int offset, imm int cpol)
typedef int v4i __attribute__((ext_vector_type(4)));
typedef __attribute__((address_space(1))) v4i as1_v4i;
typedef __attribute__((address_space(3))) v4i as3_v4i;

#if __has_builtin(__builtin_amdgcn_s_wait_asynccnt)
#define WAIT_ASYNC(n) __builtin_amdgcn_s_wait_asynccnt(n)
#else
#define WAIT_ASYNC(n) asm volatile("s_wait_asynccnt %0" ::"i"(n) : "memory")
#endif

#if __has_builtin(__builtin_amdgcn_s_wait_dscnt)
#define WAIT_DS(n) __builtin_amdgcn_s_wait_dscnt(n)
#else
#define WAIT_DS(n) asm volatile("s_wait_dscnt %0" ::"i"(n) : "memory")
#endif
#endif  // HAVE_ASYNC_LDS

namespace {

constexpr int WPB     = 8;   // waves per block (256 threads, wave32)
constexpr int SROWS   = 8;   // rows (512 B each) per pipeline stage
constexpr int NSTAGES = 12;  // 96 rows total per wave (2 tensors * 3c * 16 dzy)

// wid -> (b, z, y); lane -> x.  Row r in [0,96):
//   tensor = r<48 ? org : enh ; rr = r%48 ; c = rr>>4 ; dz = (rr>>2)&3 ; dy = rr&3
__device__ __forceinline__ const float* row_ptr(const float* __restrict__ org,
                                                const float* __restrict__ enh,
                                                int r, int b, int z, int y,
                                                int lane) {
  const float* src = (r < 48) ? org : enh;
  const int rr = (r < 48) ? r : r - 48;
  const int c  = rr >> 4;
  const int dz = (rr >> 2) & 3;
  const int dy = rr & 3;
  return src +
         ((((size_t)(b * 3 + c) * 128) + (size_t)(4 * z + dz)) * 128 +
          (size_t)(4 * y + dy)) * 128 +
         (size_t)lane * 4;
}

__global__ __launch_bounds__(256) void pool_diff_kernel(
    const float* __restrict__ org, const float* __restrict__ enh,
    float* __restrict__ g) {
  const int lane = threadIdx.x & 31;
  const int wave = threadIdx.x >> 5;
  const int wid  = blockIdx.x * WPB + wave;  // 0..4095 = ((b*32+z)*32+y)
  const int y = wid & 31;
  const int z = (wid >> 5) & 31;
  const int b = wid >> 10;

  float4 acc = make_float4(0.f, 0.f, 0.f, 0.f);

#if HAVE_ASYNC_LDS
  // 8 waves * 2 buffers * 8 rows * 512 B = 64 KB of the 320 KB WGP LDS.
  __shared__ float4 lbuf[WPB * 2 * SROWS * 32];
  float4* ldsw = lbuf + wave * (2 * SROWS * 32);

  auto issue_stage = [&](int stage, int buf) {
    const int r0 = stage * SROWS;
#pragma unroll
    for (int i = 0; i < SROWS; ++i) {
      const float* gp = row_ptr(org, enh, r0 + i, b, z, y, lane);
      float4* lp = ldsw + (buf * SROWS + i) * 32 + lane;
      // 32 lanes x 16 B = one full 512-B row per instruction, ASYNCcnt-tracked.
      __builtin_amdgcn_global_load_async_to_lds_b128((as1_v4i*)gp,
                                                     (as3_v4i*)lp, 0, 0);
    }
  };

  issue_stage(0, 0);
  for (int stage = 0; stage < NSTAGES; ++stage) {
    const int buf = stage & 1;
    if (stage + 1 < NSTAGES) {
      if (stage > 0) WAIT_DS(0);  // prior LDS reads of this buffer drained
      issue_stage(stage + 1, buf ^ 1);
      WAIT_ASYNC(SROWS);          // in-order: current stage's 8 rows landed
    } else {
      WAIT_ASYNC(0);
    }
    asm volatile("" ::: "memory");
    const float4* lb = ldsw + buf * SROWS * 32;
#pragma unroll
    for (int i = 0; i < SROWS; ++i) {
      float4 v = lb[i * 32 + lane];
      if (stage < 6) {  // org half (+), enh half (-)
        acc.x += v.x; acc.y += v.y; acc.z += v.z; acc.w += v.w;
      } else {
        acc.x -= v.x; acc.y -= v.y; acc.z -= v.z; acc.w -= v.w;
      }
    }
    asm volatile("" ::: "memory");
  }
#else
  // Fallback: direct coalesced global float4 loads (b128), no LDS staging.
#pragma unroll 4
  for (int r = 0; r < 96; ++r) {
    const float4 v = *(const float4*)row_ptr(org, enh, r, b, z, y, lane);
    if (r < 48) {
      acc.x += v.x; acc.y += v.y; acc.z += v.z; acc.w += v.w;
    } else {
      acc.x -= v.x; acc.y -= v.y; acc.z -= v.z; acc.w -= v.w;
    }
  }
#endif

  // mean over 3 channels * mean over 4x4x4 window = 1/192
  g[(size_t)wid * 32 + lane] =
      (acc.x + acc.y + acc.z + acc.w) * (1.0f / 192.0f);
}

// E[i] = sum over 6 directions of (g[i] - neighbor)^2, zero-padded borders.
__global__ __launch_bounds__(256) void stencil_kernel(
    const float* __restrict__ g, float* __restrict__ out) {
  const int idx = blockIdx.x * blockDim.x + threadIdx.x;  // 0..131071
  const int x = idx & 31;
  const int y = (idx >> 5) & 31;
  const int z = (idx >> 10) & 31;

  const float c = g[idx];
  float e = 0.f, n, d;
  n = (x > 0)  ? g[idx - 1]    : 0.f; d = c - n; e += d * d;
  n = (x < 31) ? g[idx + 1]    : 0.f; d = c - n; e += d * d;
  n = (y > 0)  ? g[idx - 32]   : 0.f; d = c - n; e += d * d;
  n = (y < 31) ? g[idx + 32]   : 0.f; d = c - n; e += d * d;
  n = (z > 0)  ? g[idx - 1024] : 0.f; d = c - n; e += d * d;
  n = (z < 31) ? g[idx + 1024] : 0.f; d = c - n; e += d * d;
  out[idx] = e;
}

}  // namespace

extern "C" void kernel_launch(void* const* d_in, const int* in_sizes, int n_in,
                              void* d_out, int out_size, void* d_ws,
                              size_t ws_size, hipStream_t stream) {
  (void)in_sizes; (void)n_in; (void)out_size; (void)ws_size;
  const float* org = (const float*)d_in[0];
  const float* enh = (const float*)d_in[1];
  float* g   = (float*)d_ws;   // 4*32*32*32 floats = 512 KB scratch
  float* out = (float*)d_out;  // 131072 floats

  // 4096 waves total: 512 blocks * 8 waves, one wave per (b,z,y) output row.
  pool_diff_kernel<<<512, 256, 0, stream>>>(org, enh, g);
  // 131072 voxels: 512 blocks * 256 threads.
  stencil_kernel<<<512, 256, 0, stream>>>(g, out);
}